// MemoryDiscriminator_24412594110620
// MI455X (gfx1250) — compile-verified
//
#include <hip/hip_runtime.h>
#include <math.h>

// ---------------------------------------------------------------------------
// MemoryDiscriminator on MI455X (gfx1250), fp32 end-to-end via
// V_WMMA_F32_16X16X4_F32 (wave32). WMMA-bound (~82 GF fp32); software-
// pipelined B-fragment loads + CDNA5 async global->LDS staging of x-block.
// ---------------------------------------------------------------------------

typedef __attribute__((ext_vector_type(2))) float v2f;
typedef __attribute__((ext_vector_type(8))) float v8f;
typedef int v4i_t __attribute__((__vector_size__(16)));
typedef __attribute__((address_space(3))) v4i_t* lds_v4i_p;

__device__ __forceinline__ v8f wmma4(v2f a, v2f b, v8f c) {
  // (neg_a, A, neg_b, B, c_mod, C, reuse_a, reuse_b)
  return __builtin_amdgcn_wmma_f32_16x16x4_f32(false, a, false, b, (short)0, c,
                                               false, false);
}
__device__ __forceinline__ float sigmoidf_(float x) {
  return 1.0f / (1.0f + expf(-x));
}
__device__ __forceinline__ float leaky02(float v) {
  return fmaxf(v, 0.2f * v);
}

#define ZV8 v8f{0.f,0.f,0.f,0.f,0.f,0.f,0.f,0.f}

// ---- CDNA5 async global->LDS copy (ASYNCcnt path), with safe fallback -----
#if __has_builtin(__builtin_amdgcn_global_load_async_to_lds_b128)
#define ASYNC_LDS 1
#endif

__device__ __forceinline__ void copy16_to_lds(const float* g, float* l) {
#ifdef ASYNC_LDS
  __builtin_amdgcn_global_load_async_to_lds_b128((v4i_t*)g, (lds_v4i_p)l, 0, 0);
#else
  *(float4*)l = *(const float4*)g;
#endif
}
__device__ __forceinline__ void wait_lds_copy() {
#ifdef ASYNC_LDS
#if __has_builtin(__builtin_amdgcn_s_wait_asynccnt)
  __builtin_amdgcn_s_wait_asynccnt(0);
#else
  asm volatile("s_wait_asynccnt 0x0" ::: "memory");
#endif
#endif
}

// ---------------------------------------------------------------------------
// K1: E = leaky(x @ W_emb^T + b_emb) tiles via WMMA; per-channel sum/sumsq
// (LDS ds_add_f32, then global atomics). B-fragments double-buffered.
// grid (32768/64, 1024/64), 128 threads (4 waves), wave = 16x64 tile.
// ---------------------------------------------------------------------------
__global__ __launch_bounds__(128) void bn_stats_kernel(
    const float* __restrict__ x, const float* __restrict__ Wemb,
    const float* __restrict__ bemb, float* __restrict__ gsum,
    float* __restrict__ gsq) {
  __shared__ float lsum[64];
  __shared__ float lsq[64];
  const int tid = threadIdx.x;
  if (tid < 64) { lsum[tid] = 0.f; lsq[tid] = 0.f; }
  __syncthreads();
  const int lane = tid & 31, w = tid >> 5;
  const int r = lane & 15, kh = (lane >> 4) & 1;
  const int m0 = blockIdx.x * 64 + 16 * w;
  const int n0 = blockIdx.y * 64;
  v8f acc[4];
#pragma unroll
  for (int s = 0; s < 4; ++s) acc[s] = ZV8;
  const float* arow = x + (size_t)(m0 + r) * 128;
  // pipelined fragment loads
  v2f ac = *(const v2f*)(arow + 2 * kh);
  v2f bc[4];
#pragma unroll
  for (int s = 0; s < 4; ++s)
    bc[s] = *(const v2f*)(Wemb + (size_t)(n0 + 16 * s + r) * 128 + 2 * kh);
  for (int k = 0; k < 128; k += 4) {
    int kn = ((k + 4 < 128) ? k + 4 : 0) + 2 * kh;
    v2f an = *(const v2f*)(arow + kn);
    v2f bnx[4];
#pragma unroll
    for (int s = 0; s < 4; ++s)
      bnx[s] = *(const v2f*)(Wemb + (size_t)(n0 + 16 * s + r) * 128 + kn);
#pragma unroll
    for (int s = 0; s < 4; ++s) acc[s] = wmma4(ac, bc[s], acc[s]);
    ac = an;
#pragma unroll
    for (int s = 0; s < 4; ++s) bc[s] = bnx[s];
  }
#pragma unroll
  for (int s = 0; s < 4; ++s) {
    int cl = 16 * s + r;
    float bb = bemb[n0 + cl];
    float ps = 0.f, pq = 0.f;
#pragma unroll
    for (int i = 0; i < 8; ++i) {
      float v = leaky02(acc[s][i] + bb);
      ps += v;
      pq += v * v;
    }
    atomicAdd(&lsum[cl], ps);
    atomicAdd(&lsq[cl], pq);
  }
  __syncthreads();
  if (tid < 64) {
    atomicAdd(&gsum[n0 + tid], lsum[tid]);
    atomicAdd(&gsq[n0 + tid], lsq[tid]);
  }
}

// ---------------------------------------------------------------------------
// K2: finalize BN
// ---------------------------------------------------------------------------
__global__ void bn_finalize_kernel(const float* __restrict__ gsum,
                                   const float* __restrict__ gsq,
                                   const float* __restrict__ gamma,
                                   const float* __restrict__ beta,
                                   float* __restrict__ scale,
                                   float* __restrict__ shift) {
  int c = blockIdx.x * blockDim.x + threadIdx.x;
  if (c >= 1024) return;
  const float inv = 1.0f / 32768.0f;
  float mean = gsum[c] * inv;
  float var = gsq[c] * inv - mean * mean;
  float sc = gamma[c] * rsqrtf(var + 1e-5f);
  scale[c] = sc;
  shift[c] = beta[c] - mean * sc;
}

// ---------------------------------------------------------------------------
// K3: Gx = BN(leaky(x@W_emb^T + b)) @ w_ih_x^T + b_ih_x, fused.
// WG = 16 waves, M-block = 64, full N strip = 768.
// x-block (64x128) async-staged to LDS once; 16 H-chunks of 64:
//   (a) WMMA-recompute E chunk into LDS (stride 68 -> conflict free)
//   (b) WMMA-accumulate 12 frags/wave, B double-buffered in registers.
// LDS = 33.8KB (xs) + 17.4KB (E) = 51KB.
// ---------------------------------------------------------------------------
#define XSTR 132
#define ESTRIDE 68
__global__ __launch_bounds__(512) void gx_kernel(
    const float* __restrict__ x, const float* __restrict__ Wemb,
    const float* __restrict__ bemb, const float* __restrict__ scale,
    const float* __restrict__ shift, const float* __restrict__ wih,
    const float* __restrict__ bih, float* __restrict__ Gx) {
  __shared__ float xs[64 * XSTR];
  __shared__ float E[64 * ESTRIDE];
  const int tid = threadIdx.x, lane = tid & 31, w = tid >> 5;
  const int r = lane & 15, kh = (lane >> 4) & 1;
  const int mblk = blockIdx.x * 64;
  const int mt = w & 3;   // m-tile (both phases)
  const int wg = w >> 2;  // 0..3
  // ---- stage x block (rows mblk..mblk+63) into LDS via async copies ----
#pragma unroll
  for (int rr = 0; rr < 4; ++rr) {
    int row = w + 16 * rr;
    copy16_to_lds(x + (size_t)(mblk + row) * 128 + lane * 4,
                  &xs[row * XSTR + lane * 4]);
  }
  wait_lds_copy();
  __syncthreads();

  v8f acc[12];
#pragma unroll
  for (int j = 0; j < 12; ++j) acc[j] = ZV8;

  for (int hc = 0; hc < 16; ++hc) {
    // Phase A: E chunk (64 rows x 64 channels), 1 subtile per wave (mt, wg)
    {
      v8f e = ZV8;
      const float* arow = &xs[(16 * mt + r) * XSTR];
      const float* brow = Wemb + (size_t)(hc * 64 + 16 * wg + r) * 128;
      v2f ac = *(const v2f*)(arow + 2 * kh);
      v2f bc = *(const v2f*)(brow + 2 * kh);
      for (int k = 0; k < 128; k += 4) {
        int kn = ((k + 4 < 128) ? k + 4 : 0) + 2 * kh;
        v2f an = *(const v2f*)(arow + kn);
        v2f bn2 = *(const v2f*)(brow + kn);
        e = wmma4(ac, bc, e);
        ac = an;
        bc = bn2;
      }
      int cl = 16 * wg + r;
      int cg = hc * 64 + cl;
      float bb = bemb[cg], sc = scale[cg], sh = shift[cg];
#pragma unroll
      for (int i = 0; i < 8; ++i) {
        float v = leaky02(e[i] + bb);
        E[(16 * mt + i + 8 * kh) * ESTRIDE + cl] = v * sc + sh;
      }
    }
    __syncthreads();
    // Phase B: accumulate Gx frags from LDS E chunk, B double-buffered
    const float* ea = &E[(16 * mt + r) * ESTRIDE];
    const float* wbase = wih + (size_t)hc * 64;
    if (hc < 15) __builtin_prefetch(wih + (size_t)(hc + 1) * 64, 0, 1);
    v2f bc[12];
#pragma unroll
    for (int j = 0; j < 12; ++j)
      bc[j] =
          *(const v2f*)(wbase + (size_t)(16 * (wg + 4 * j) + r) * 1024 + 2 * kh);
    for (int k = 0; k < 64; k += 4) {
      int kn = ((k + 4 < 64) ? k + 4 : 0) + 2 * kh;
      v2f a = *(const v2f*)(ea + k + 2 * kh);
      v2f bnx[12];
#pragma unroll
      for (int j = 0; j < 12; ++j)
        bnx[j] =
            *(const v2f*)(wbase + (size_t)(16 * (wg + 4 * j) + r) * 1024 + kn);
#pragma unroll
      for (int j = 0; j < 12; ++j) acc[j] = wmma4(a, bc[j], acc[j]);
#pragma unroll
      for (int j = 0; j < 12; ++j) bc[j] = bnx[j];
    }
    __syncthreads();
  }
#pragma unroll
  for (int j = 0; j < 12; ++j) {
    int col = 16 * (wg + 4 * j) + r;
    float bb = bih[col];
#pragma unroll
    for (int i = 0; i < 8; ++i)
      Gx[(size_t)(mblk + 16 * mt + i + 8 * kh) * 768 + col] = acc[j][i] + bb;
  }
}

// ---------------------------------------------------------------------------
// K4: generic C(MxN) = A(MxK) @ W(NxK)^T + bias via WMMA (used for Gm).
// ---------------------------------------------------------------------------
__global__ __launch_bounds__(128) void gemm_xwT_kernel(
    const float* __restrict__ A, const float* __restrict__ W,
    const float* __restrict__ bias, float* __restrict__ C, int M, int N,
    int K) {
  const int tid = threadIdx.x, lane = tid & 31, w = tid >> 5;
  const int r = lane & 15, kh = (lane >> 4) & 1;
  const int m0 = blockIdx.x * 64 + 16 * w;
  const int n0 = blockIdx.y * 64;
  v8f acc[4];
#pragma unroll
  for (int s = 0; s < 4; ++s) acc[s] = ZV8;
  const float* arow = A + (size_t)(m0 + r) * K;
  v2f ac = *(const v2f*)(arow + 2 * kh);
  v2f bc[4];
#pragma unroll
  for (int s = 0; s < 4; ++s)
    bc[s] = *(const v2f*)(W + (size_t)(n0 + 16 * s + r) * K + 2 * kh);
  for (int k = 0; k < K; k += 4) {
    int kn = ((k + 4 < K) ? k + 4 : 0) + 2 * kh;
    v2f an = *(const v2f*)(arow + kn);
    v2f bnx[4];
#pragma unroll
    for (int s = 0; s < 4; ++s)
      bnx[s] = *(const v2f*)(W + (size_t)(n0 + 16 * s + r) * K + kn);
#pragma unroll
    for (int s = 0; s < 4; ++s) acc[s] = wmma4(ac, bc[s], acc[s]);
    ac = an;
#pragma unroll
    for (int s = 0; s < 4; ++s) bc[s] = bnx[s];
  }
#pragma unroll
  for (int s = 0; s < 4; ++s) {
    int col = n0 + 16 * s + r;
    float bb = bias[col];
#pragma unroll
    for (int i = 0; i < 8; ++i)
      C[(size_t)(m0 + i + 8 * kh) * N + col] = acc[s][i] + bb;
  }
}

// ---------------------------------------------------------------------------
// K5: hm recurrence (memory is batch-broadcast -> single 256-vector GRU).
// ---------------------------------------------------------------------------
__global__ __launch_bounds__(256) void hm_kernel(const float* __restrict__ Gm,
                                                 const float* __restrict__ whh,
                                                 const float* __restrict__ bhh,
                                                 float* __restrict__ hm_out) {
  __shared__ float hm[256];
  const int j = threadIdx.x;
  hm[j] = 0.f;
  __syncthreads();
  const float br = bhh[j], bz = bhh[256 + j], bn = bhh[512 + j];
  const float* wr = whh + (size_t)j * 256;
  const float* wz = whh + (size_t)(256 + j) * 256;
  const float* wn = whh + (size_t)(512 + j) * 256;
  for (int t = 0; t < 128; ++t) {
    float a0 = 0.f, a1 = 0.f, a2 = 0.f;
    for (int k = 0; k < 256; ++k) {
      float hv = hm[k];
      a0 += wr[k] * hv;
      a1 += wz[k] * hv;
      a2 += wn[k] * hv;
    }
    const float* g = Gm + (size_t)t * 768;
    float rr = sigmoidf_(g[j] + a0 + br);
    float zz = sigmoidf_(g[256 + j] + a1 + bz);
    float nn = tanhf(g[512 + j] + rr * (a2 + bn));
    float hnew = (1.f - zz) * nn + zz * hm[j];
    __syncthreads();
    hm[j] = hnew;
    __syncthreads();
  }
  hm_out[j] = hm[j];
}

// ---------------------------------------------------------------------------
// K6: hx recurrence. 16 WGs x 512 threads (16 waves = 4/SIMD for L2 latency
// hiding); each WG owns 16 batch rows, hx transposed in LDS. Per step:
// Hg(16x768) = hx @ w_hh_x^T via WMMA (K=256, 3 N-tiles/wave, B frags
// double-buffered), then GRU elementwise update vs precomputed Gx gates.
// ---------------------------------------------------------------------------
__global__ __launch_bounds__(512) void hx_kernel(const float* __restrict__ Gx,
                                                 const float* __restrict__ whh,
                                                 const float* __restrict__ bhh,
                                                 float* __restrict__ hx_out) {
  __shared__ float hxT[256 * 16];  // [col][row], 16 KB
  __shared__ float Hg[16 * 768];   // 48 KB
  const int tid = threadIdx.x, lane = tid & 31, w = tid >> 5;  // w: 0..15
  const int r = lane & 15, kh = (lane >> 4) & 1;
  const int m0g = blockIdx.x * 16;
  for (int idx = tid; idx < 4096; idx += 512) hxT[idx] = 0.f;
  __syncthreads();

  for (int t = 0; t < 128; ++t) {
    v8f acc[3];
#pragma unroll
    for (int j = 0; j < 3; ++j) acc[j] = ZV8;
    v2f bc[3];
#pragma unroll
    for (int j = 0; j < 3; ++j)
      bc[j] = *(const v2f*)(whh + (size_t)(16 * (w + 16 * j) + r) * 256 +
                            2 * kh);
    for (int k = 0; k < 256; k += 4) {
      int kc = k + 2 * kh;
      int kn = ((k + 4 < 256) ? k + 4 : 0) + 2 * kh;
      v2f a = {hxT[kc * 16 + r], hxT[(kc + 1) * 16 + r]};
      v2f bnx[3];
#pragma unroll
      for (int j = 0; j < 3; ++j)
        bnx[j] = *(const v2f*)(whh + (size_t)(16 * (w + 16 * j) + r) * 256 + kn);
#pragma unroll
      for (int j = 0; j < 3; ++j) acc[j] = wmma4(a, bc[j], acc[j]);
#pragma unroll
      for (int j = 0; j < 3; ++j) bc[j] = bnx[j];
    }
#pragma unroll
    for (int j = 0; j < 3; ++j) {
      int col = 16 * (w + 16 * j) + r;
#pragma unroll
      for (int i = 0; i < 8; ++i) Hg[(i + 8 * kh) * 768 + col] = acc[j][i];
    }
    __syncthreads();
    for (int idx = tid; idx < 4096; idx += 512) {
      int row = idx >> 8, o = idx & 255;
      size_t gb = ((size_t)(m0g + row) * 128 + t) * 768;
      float gr = Gx[gb + o], gz = Gx[gb + 256 + o], gn = Gx[gb + 512 + o];
      float hr = Hg[row * 768 + o] + bhh[o];
      float hz = Hg[row * 768 + 256 + o] + bhh[256 + o];
      float hn = Hg[row * 768 + 512 + o] + bhh[512 + o];
      float rr = sigmoidf_(gr + hr);
      float zz = sigmoidf_(gz + hz);
      float nn = tanhf(gn + rr * hn);
      float hold = hxT[o * 16 + row];
      hxT[o * 16 + row] = (1.f - zz) * nn + zz * hold;
    }
    __syncthreads();
  }
  for (int idx = tid; idx < 4096; idx += 512) {
    int row = idx >> 8, o = idx & 255;
    hx_out[(size_t)(m0g + row) * 256 + o] = hxT[o * 16 + row];
  }
}

// ---------------------------------------------------------------------------
// K7: cosine-sim gate (SIM=4) + blended output of last timestep.
// ---------------------------------------------------------------------------
__global__ __launch_bounds__(256) void out_kernel(
    const float* __restrict__ hxF, const float* __restrict__ hmv,
    const float* __restrict__ Wsx, const float* __restrict__ bsx,
    const float* __restrict__ Wsm, const float* __restrict__ bsm,
    float* __restrict__ out) {
  __shared__ float gL[256];
  __shared__ float hmL[256];
  const int tid = threadIdx.x;
  hmL[tid] = hmv[tid];
  __syncthreads();
  float qx[4], qm[4];
  const float* hx = hxF + (size_t)tid * 256;
#pragma unroll
  for (int s = 0; s < 4; ++s) {
    float ax = bsx[s], am = bsm[s];
    const float* wx = Wsx + s * 256;
    const float* wm = Wsm + s * 256;
    for (int k = 0; k < 256; ++k) {
      ax += wx[k] * hx[k];
      am += wm[k] * hmL[k];
    }
    qx[s] = ax;
    qm[s] = am;
  }
  float num = 0.f, nx = 0.f, nm = 0.f;
#pragma unroll
  for (int s = 0; s < 4; ++s) {
    num += qx[s] * qm[s];
    nx += qx[s] * qx[s];
    nm += qm[s] * qm[s];
  }
  float den = fmaxf(sqrtf(nx), 1e-8f) * fmaxf(sqrtf(nm), 1e-8f);
  gL[tid] = sigmoidf_(num / den);
  __syncthreads();
  for (int idx = tid; idx < 65536; idx += 256) {
    int row = idx >> 8, o = idx & 255;
    float g = gL[row];
    out[idx] = g * hxF[(size_t)row * 256 + o] + (1.f - g) * hmL[o];
  }
}

// ---------------------------------------------------------------------------
extern "C" void kernel_launch(void* const* d_in, const int* in_sizes, int n_in,
                              void* d_out, int out_size, void* d_ws,
                              size_t ws_size, hipStream_t stream) {
  const float* x = (const float*)d_in[0];
  const float* Wemb = (const float*)d_in[1];
  const float* bemb = (const float*)d_in[2];
  const float* gamma = (const float*)d_in[3];
  const float* beta = (const float*)d_in[4];
  const float* mem = (const float*)d_in[5];  // (1,T,H) -> (128,1024)
  const float* wihx = (const float*)d_in[6];
  const float* whhx = (const float*)d_in[7];
  const float* bihx = (const float*)d_in[8];
  const float* bhhx = (const float*)d_in[9];
  const float* wihm = (const float*)d_in[10];
  const float* whhm = (const float*)d_in[11];
  const float* bihm = (const float*)d_in[12];
  const float* bhhm = (const float*)d_in[13];
  const float* Wsx = (const float*)d_in[14];
  const float* bsx = (const float*)d_in[15];
  const float* Wsm = (const float*)d_in[16];
  const float* bsm = (const float*)d_in[17];
  float* out = (float*)d_out;

  float* ws = (float*)d_ws;
  float* gsum = ws;                      // 1024
  float* gsq = ws + 1024;                // 1024
  float* scale = ws + 2048;              // 1024
  float* shift = ws + 3072;              // 1024
  float* Gm = ws + 4096;                 // 128*768
  float* hmv = Gm + 128 * 768;           // 256
  float* hxF = hmv + 256;                // 256*256
  float* Gx = hxF + 65536;               // 32768*768 (~96 MB total ws use)

  (void)hipMemsetAsync(gsum, 0, 2048 * sizeof(float), stream);
  bn_stats_kernel<<<dim3(512, 16), 128, 0, stream>>>(x, Wemb, bemb, gsum, gsq);
  bn_finalize_kernel<<<4, 256, 0, stream>>>(gsum, gsq, gamma, beta, scale,
                                            shift);
  gx_kernel<<<512, 512, 0, stream>>>(x, Wemb, bemb, scale, shift, wihx, bihx,
                                     Gx);
  gemm_xwT_kernel<<<dim3(2, 12), 128, 0, stream>>>(mem, wihm, bihm, Gm, 128,
                                                   768, 1024);
  hm_kernel<<<1, 256, 0, stream>>>(Gm, whhm, bhhm, hmv);
  hx_kernel<<<16, 512, 0, stream>>>(Gx, whhx, bhhx, hxF);
  out_kernel<<<1, 256, 0, stream>>>(hxF, hmv, Wsx, bsx, Wsm, bsm, out);
}